// EncoderModel_27015344292445
// MI455X (gfx1250) — compile-verified
//
#include <hip/hip_runtime.h>

typedef __attribute__((ext_vector_type(8))) float v8f;
typedef __attribute__((ext_vector_type(2))) float v2f;

// ---------------------------------------------------------------------------
// Fusion: out[i] = concat(embed[node_index[i]], x[i] @ W.T + b)  -> [n, 32]
// One thread per (node, output-feature). fin = 5 (user) or 2 (course).
// ---------------------------------------------------------------------------
__global__ void fuse_kernel(const float* __restrict__ x,
                            const float* __restrict__ embed,
                            const int* __restrict__ node_index,
                            const float* __restrict__ W,
                            const float* __restrict__ b,
                            float* __restrict__ out,
                            int n, int fin) {
  int t = blockIdx.x * blockDim.x + threadIdx.x;
  int node = t >> 5;
  int j = t & 31;
  if (node >= n) return;
  float v;
  if (j < 16) {
    v = embed[(long long)node_index[node] * 16 + j];
  } else {
    int r = j - 16;
    float acc = b[r];
    const float* xr = x + (long long)node * fin;
    for (int k = 0; k < fin; ++k) acc = fmaf(xr[k], W[r * fin + k], acc);
    v = acc;
  }
  out[(long long)node * 32 + j] = v;
}

// ---------------------------------------------------------------------------
// Degree count: deg[idx[e]] += 1  (float, matching reference semantics)
// ---------------------------------------------------------------------------
__global__ void degree_kernel(const int* __restrict__ idx,
                              float* __restrict__ deg, int E) {
  int t = blockIdx.x * blockDim.x + threadIdx.x;
  if (t < E) atomicAdd(&deg[idx[t]], 1.0f);
}

// ---------------------------------------------------------------------------
// Edge scatter-add: agg[scatter_idx[e], :] += srcf[gather_idx[e], :]  (F = 32)
// 8 threads per edge; each thread reads one float4 and issues 4 f32 atomics.
// Gather tables and atomic targets are L2-resident (<=64 MB vs 192 MB L2).
// ---------------------------------------------------------------------------
__global__ void scatter_kernel(const float* __restrict__ srcf,
                               const int* __restrict__ gather_idx,
                               const int* __restrict__ scatter_idx,
                               float* __restrict__ agg,
                               int E) {
  int t = blockIdx.x * blockDim.x + threadIdx.x;
  int e = t >> 3;
  int part = t & 7;
  if (e >= E) return;
  int s = gather_idx[e];
  int d = scatter_idx[e];
  float4 v = ((const float4*)(srcf + (long long)s * 32))[part];
  float* o = agg + (long long)d * 32 + part * 4;
  atomicAdd(o + 0, v.x);
  atomicAdd(o + 1, v.y);
  atomicAdd(o + 2, v.z);
  atomicAdd(o + 3, v.w);
}

// ---------------------------------------------------------------------------
// SAGE finalize with exact-f32 WMMA (V_WMMA_F32_16X16X4_F32):
//   out[i, :] = act( (agg[i,:]/max(deg[i],1)) @ Wl.T + bl + xdst[i,:] @ Wr.T )
//
// One wave32 per 16-node tile. K = 32 is consumed as 8 chained K=4 WMMAs for
// the aggregate term plus 8 for the root term (per 16-wide N tile).
//
// Register layouts per CDNA5 ISA 7.12.2 (32-bit element):
//   A (16x4 f32): lane l -> row m = l&15; VGPR0 K = 2*(l>>4), VGPR1 K+1
//   B (4x16 f32): lane l -> col n = l&15; VGPR0 K = 2*(l>>4), VGPR1 K+1
//   C/D (16x16 f32): lane l -> col n = l&15; vgpr v -> row m = v + 8*(l>>4)
// ---------------------------------------------------------------------------
template <int NTILES, bool RELU>
__global__ void sage_wmma_kernel(const float* __restrict__ agg,
                                 const float* __restrict__ deg,
                                 const float* __restrict__ xdst,
                                 const float* __restrict__ Wl,
                                 const float* __restrict__ bl,
                                 const float* __restrict__ Wr,
                                 float* __restrict__ out,
                                 int n) {
  const int OUTF = NTILES * 16;
  int lane = threadIdx.x & 31;
  int wave = (blockIdx.x * blockDim.x + threadIdx.x) >> 5;
  int nb = wave * 16;
  if (nb >= n) return;                 // wave-uniform: EXEC stays all-ones
  int m  = lane & 15;
  int hi = lane >> 4;

  int node = nb + m;
  bool nv = node < n;
  int ncl = nv ? node : (n - 1);       // clamp so speculative loads stay in-bounds
  float dinv = nv ? (1.0f / fmaxf(deg[ncl], 1.0f)) : 0.0f;
  const float* arow = agg  + (long long)ncl * 32;
  const float* xrow = xdst + (long long)ncl * 32;

  // A fragments: step s covers K in [4s, 4s+4); this lane holds K = 4s+2*hi, +1.
  v2f a_agg[8], a_x[8];
#pragma unroll
  for (int s = 0; s < 8; ++s) {
    int k = 4 * s + 2 * hi;
    a_agg[s].x = arow[k] * dinv;       // dinv == 0 for padded lanes -> 0
    a_agg[s].y = arow[k + 1] * dinv;
    a_x[s].x   = nv ? xrow[k]     : 0.0f;
    a_x[s].y   = nv ? xrow[k + 1] : 0.0f;
  }

  bool full = (nb + 16 <= n);          // wave-uniform fast path (always true here)

#pragma unroll
  for (int t = 0; t < NTILES; ++t) {
    int ng = t * 16 + m;               // global output-feature index (B column)
    const float* wl = Wl + ng * 32;
    const float* wr = Wr + ng * 32;
    float bias = bl[ng];
    v8f c;
#pragma unroll
    for (int v = 0; v < 8; ++v) c[v] = bias;

#pragma unroll
    for (int s = 0; s < 8; ++s) {
      int k = 4 * s + 2 * hi;
      v2f b;
      b.x = wl[k];
      b.y = wl[k + 1];
      c = __builtin_amdgcn_wmma_f32_16x16x4_f32(false, a_agg[s], false, b,
                                                (short)0, c, false, false);
    }
#pragma unroll
    for (int s = 0; s < 8; ++s) {
      int k = 4 * s + 2 * hi;
      v2f b;
      b.x = wr[k];
      b.y = wr[k + 1];
      c = __builtin_amdgcn_wmma_f32_16x16x4_f32(false, a_x[s], false, b,
                                                (short)0, c, false, false);
    }

    if (full) {                        // branch-free store path
#pragma unroll
      for (int v = 0; v < 8; ++v) {
        float val = c[v];
        if (RELU) val = fmaxf(val, 0.0f);
        out[(long long)(nb + v + 8 * hi) * OUTF + ng] = val;
      }
    } else {                           // tail wave only
#pragma unroll
      for (int v = 0; v < 8; ++v) {
        int nd = nb + v + 8 * hi;
        if (nd < n) {
          float val = c[v];
          if (RELU) val = fmaxf(val, 0.0f);
          out[(long long)nd * OUTF + ng] = val;
        }
      }
    }
  }
}

// ---------------------------------------------------------------------------
// Decode: out[l] = dot(ou[label_src[l]], oc[label_dst[l]]) over 16 dims
// ---------------------------------------------------------------------------
__global__ void decode_kernel(const float* __restrict__ ou,
                              const float* __restrict__ oc,
                              const int* __restrict__ ls,
                              const int* __restrict__ ld,
                              float* __restrict__ out, int L) {
  int t = blockIdx.x * blockDim.x + threadIdx.x;
  if (t >= L) return;
  const float4* a = (const float4*)(ou + (long long)ls[t] * 16);
  const float4* b = (const float4*)(oc + (long long)ld[t] * 16);
  float acc = 0.0f;
#pragma unroll
  for (int i = 0; i < 4; ++i) {
    float4 xa = a[i], xb = b[i];
    acc += xa.x * xb.x + xa.y * xb.y + xa.z * xb.z + xa.w * xb.w;
  }
  out[t] = acc;
}

// ---------------------------------------------------------------------------
extern "C" void kernel_launch(void* const* d_in, const int* in_sizes, int n_in,
                              void* d_out, int out_size, void* d_ws, size_t ws_size,
                              hipStream_t stream) {
  const float* user_x       = (const float*)d_in[0];
  const float* course_x     = (const float*)d_in[1];
  const float* user_embed   = (const float*)d_in[2];
  const float* course_embed = (const float*)d_in[3];
  const float* Wu = (const float*)d_in[4];
  const float* bu = (const float*)d_in[5];
  const float* Wc = (const float*)d_in[6];
  const float* bc = (const float*)d_in[7];
  const float* c1e_Wl = (const float*)d_in[8];
  const float* c1e_bl = (const float*)d_in[9];
  const float* c1e_Wr = (const float*)d_in[10];
  const float* c1r_Wl = (const float*)d_in[11];
  const float* c1r_bl = (const float*)d_in[12];
  const float* c1r_Wr = (const float*)d_in[13];
  const float* c2e_Wl = (const float*)d_in[14];
  const float* c2e_bl = (const float*)d_in[15];
  const float* c2e_Wr = (const float*)d_in[16];
  const float* c2r_Wl = (const float*)d_in[17];
  const float* c2r_bl = (const float*)d_in[18];
  const float* c2r_Wr = (const float*)d_in[19];
  const int* user_idx   = (const int*)d_in[20];
  const int* course_idx = (const int*)d_in[21];
  const int* edge_src   = (const int*)d_in[22];
  const int* edge_dst   = (const int*)d_in[23];
  const int* label_src  = (const int*)d_in[24];
  const int* label_dst  = (const int*)d_in[25];
  float* outp = (float*)d_out;

  const int NU = in_sizes[2] / 16;   // user_embed  [NU,16]
  const int NC = in_sizes[3] / 16;   // course_embed[NC,16]
  const int E  = in_sizes[22];
  const int L  = in_sizes[24];

  // Carve workspace (~229 MB total)
  char* p = (char*)d_ws;
  auto carve = [&](size_t bytes) -> float* {
    float* r = (float*)p;
    p += (bytes + 255) & ~((size_t)255);
    return r;
  };
  float* xu   = carve((size_t)NU * 32 * 4);
  float* xc   = carve((size_t)NC * 32 * 4);
  float* aggu = carve((size_t)NU * 32 * 4);
  float* aggc = carve((size_t)NC * 32 * 4);
  float* degu = carve((size_t)NU * 4);
  float* degc = carve((size_t)NC * 4);
  float* hu   = carve((size_t)NU * 32 * 4);
  float* hc   = carve((size_t)NC * 32 * 4);
  float* ou   = carve((size_t)NU * 16 * 4);
  float* oc   = carve((size_t)NC * 16 * 4);

  const int B = 256;
  auto nblk = [](long long t, int b) { return (unsigned)((t + b - 1) / b); };

  // Zero accumulators and degrees (memset nodes are graph-capture safe)
  hipMemsetAsync(aggu, 0, (size_t)NU * 32 * 4, stream);
  hipMemsetAsync(aggc, 0, (size_t)NC * 32 * 4, stream);
  hipMemsetAsync(degu, 0, (size_t)NU * 4, stream);
  hipMemsetAsync(degc, 0, (size_t)NC * 4, stream);

  // Feature fusion
  fuse_kernel<<<nblk((long long)NU * 32, B), B, 0, stream>>>(
      user_x, user_embed, user_idx, Wu, bu, xu, NU, 5);
  fuse_kernel<<<nblk((long long)NC * 32, B), B, 0, stream>>>(
      course_x, course_embed, course_idx, Wc, bc, xc, NC, 2);

  // Degrees (shared by both layers)
  degree_kernel<<<nblk(E, B), B, 0, stream>>>(edge_dst, degc, E);
  degree_kernel<<<nblk(E, B), B, 0, stream>>>(edge_src, degu, E);

  // Layer 1: scatter-add then WMMA finalize (+relu), hidden dim 32
  scatter_kernel<<<nblk((long long)E * 8, B), B, 0, stream>>>(xu, edge_src, edge_dst, aggc, E);
  scatter_kernel<<<nblk((long long)E * 8, B), B, 0, stream>>>(xc, edge_dst, edge_src, aggu, E);
  sage_wmma_kernel<2, true><<<nblk(((long long)(NC + 15) / 16) * 32, B), B, 0, stream>>>(
      aggc, degc, xc, c1e_Wl, c1e_bl, c1e_Wr, hc, NC);
  sage_wmma_kernel<2, true><<<nblk(((long long)(NU + 15) / 16) * 32, B), B, 0, stream>>>(
      aggu, degu, xu, c1r_Wl, c1r_bl, c1r_Wr, hu, NU);

  // Layer 2: re-zero, scatter-add, WMMA finalize (no relu), output dim 16
  hipMemsetAsync(aggu, 0, (size_t)NU * 32 * 4, stream);
  hipMemsetAsync(aggc, 0, (size_t)NC * 32 * 4, stream);
  scatter_kernel<<<nblk((long long)E * 8, B), B, 0, stream>>>(hu, edge_src, edge_dst, aggc, E);
  scatter_kernel<<<nblk((long long)E * 8, B), B, 0, stream>>>(hc, edge_dst, edge_src, aggu, E);
  sage_wmma_kernel<1, false><<<nblk(((long long)(NC + 15) / 16) * 32, B), B, 0, stream>>>(
      aggc, degc, hc, c2e_Wl, c2e_bl, c2e_Wr, oc, NC);
  sage_wmma_kernel<1, false><<<nblk(((long long)(NU + 15) / 16) * 32, B), B, 0, stream>>>(
      aggu, degu, hu, c2r_Wl, c2r_bl, c2r_Wr, ou, NU);

  // Decode
  decode_kernel<<<nblk(L, B), B, 0, stream>>>(ou, oc, label_src, label_dst, outp, L);
}